// decomp_layer_69810398429229
// MI455X (gfx1250) — compile-verified
//
#include <hip/hip_runtime.h>
#include <stdint.h>

// Problem sizes from the reference
#define B_C 2
#define N_C 262144
#define E_C 128                    // floats per row (512 bytes)
#define K_C 4
#define WPB 8                      // waves per block (wave32 -> 256 threads)
#define TPB (WPB * 32)
#define ROW_BYTES (E_C * 4)        // 512
#define GRP_BYTES (K_C * ROW_BYTES)// 2048 (one staging buffer)

typedef float v4f __attribute__((ext_vector_type(4)));

// ---- CDNA5 async-tensor path: ASYNCcnt-tracked gather staging into LDS ----
// GVS mode: mem_addr = SADDR(64) + VADDR(32) ; each active lane moves 16B.
// NT variant: stream level-0 x through without displacing L2 (keeps means0 resident).
template <bool SRC_NT>
__device__ __forceinline__ void async_row_b128(uint32_t lds_byte_addr,
                                               uint32_t glb_byte_off,
                                               uint64_t sbase) {
  if (SRC_NT) {
    asm volatile("global_load_async_to_lds_b128 %0, %1, %2 th:TH_LOAD_NT"
                 :: "v"(lds_byte_addr), "v"(glb_byte_off), "s"(sbase)
                 : "memory");
  } else {
    asm volatile("global_load_async_to_lds_b128 %0, %1, %2"
                 :: "v"(lds_byte_addr), "v"(glb_byte_off), "s"(sbase)
                 : "memory");
  }
}
#define S_WAIT_ASYNCCNT_4() asm volatile("s_wait_asynccnt 0x4" ::: "memory")
#define S_WAIT_ASYNCCNT_0() asm volatile("s_wait_asynccnt 0x0" ::: "memory")

// One level: src (B,Ns,E) gathered by idx (Ng,4) -> detail (B,Ng*4,E), means (B,Ng,E)
template <bool SRC_NT>
__global__ __launch_bounds__(TPB) void decomp_level_kernel(
    const float* __restrict__ src, const int* __restrict__ idx,
    float* __restrict__ detail, float* __restrict__ means,
    int Ns, int Ng)
{
  __shared__ __align__(16) float stage[WPB * 2 * K_C * E_C]; // 32 KB/block, dbl-buffered per wave

  const int lane  = threadIdx.x & 31;
  const int wave  = threadIdx.x >> 5;
  const int gwave = blockIdx.x * WPB + wave;
  const int nwav  = gridDim.x * WPB;
  const int total = B_C * Ng;                 // groups across the batch
  if (gwave >= total) return;                 // wave-uniform exit, EXEC stays full

  float* wbuf = &stage[(size_t)wave * 2 * K_C * E_C];
  const uint32_t lbase = (uint32_t)(uintptr_t)wbuf;   // wave-relative LDS byte addr (low 32b)
  const uint64_t sbase = (uint64_t)(uintptr_t)src;
  const uint32_t laneB = (uint32_t)lane * 16u;

  auto issue = [&](int t, uint32_t buf_lds) {
    const int b = (t >= Ng) ? 1 : 0;
    const int g = t - b * Ng;
    const int4 iv = ((const int4*)idx)[g];            // uniform 16B index fetch
    const uint32_t boff = (uint32_t)b * (uint32_t)Ns * (uint32_t)ROW_BYTES;
    async_row_b128<SRC_NT>(buf_lds + 0u*ROW_BYTES + laneB, boff + (uint32_t)iv.x * ROW_BYTES + laneB, sbase);
    async_row_b128<SRC_NT>(buf_lds + 1u*ROW_BYTES + laneB, boff + (uint32_t)iv.y * ROW_BYTES + laneB, sbase);
    async_row_b128<SRC_NT>(buf_lds + 2u*ROW_BYTES + laneB, boff + (uint32_t)iv.z * ROW_BYTES + laneB, sbase);
    async_row_b128<SRC_NT>(buf_lds + 3u*ROW_BYTES + laneB, boff + (uint32_t)iv.w * ROW_BYTES + laneB, sbase);
  };

  issue(gwave, lbase);                                // prologue into buffer 0
  uint32_t parity = 0;

  for (int t = gwave; t < total; t += nwav) {
    const int tn = t + nwav;
    const uint32_t curOff = parity ? (uint32_t)GRP_BYTES : 0u;

    const int tn2 = t + 2 * nwav;                     // prefetch index stream ahead
    if (tn2 < total) {
      const int b2 = (tn2 >= Ng) ? 1 : 0;
      __builtin_prefetch(&idx[(size_t)(tn2 - b2 * Ng) * 4], 0, 0);
    }

    if (tn < total) {
      issue(tn, lbase + (curOff ^ (uint32_t)GRP_BYTES));
      S_WAIT_ASYNCCNT_4();      // async loads complete in order -> current 4 done
    } else {
      S_WAIT_ASYNCCNT_0();
    }

    const float* buf = (const float*)((const char*)wbuf + curOff);
    const v4f s0 = *(const v4f*)&buf[0 * E_C + lane * 4];
    const v4f s1 = *(const v4f*)&buf[1 * E_C + lane * 4];
    const v4f s2 = *(const v4f*)&buf[2 * E_C + lane * 4];
    const v4f s3 = *(const v4f*)&buf[3 * E_C + lane * 4];

    const v4f m = (s0 + s1 + s2 + s3) * 0.25f;

    const int b = (t >= Ng) ? 1 : 0;
    const int g = t - b * Ng;
    float* dp = detail + (((size_t)b * Ng + g) * 4) * E_C + lane * 4;
    // detail is write-once, never re-read: stream past L2 (TH_STORE_NT)
    __builtin_nontemporal_store(s0 - m, (v4f*)(dp + 0 * E_C));
    __builtin_nontemporal_store(s1 - m, (v4f*)(dp + 1 * E_C));
    __builtin_nontemporal_store(s2 - m, (v4f*)(dp + 2 * E_C));
    __builtin_nontemporal_store(s3 - m, (v4f*)(dp + 3 * E_C));

    // means feed the next level: regular-temporal so they stay in the 192MB L2
    *(v4f*)(means + ((size_t)b * Ng + g) * E_C + lane * 4) = m;

    parity ^= 1u;
  }
}

static int blocks_for(long total_groups) {
  long need = (total_groups + WPB - 1) / WPB;
  if (need > 4096) need = 4096;     // cap so waves loop -> double buffering engages
  return (int)need;
}

extern "C" void kernel_launch(void* const* d_in, const int* in_sizes, int n_in,
                              void* d_out, int out_size, void* d_ws, size_t ws_size,
                              hipStream_t stream) {
  (void)in_sizes; (void)n_in; (void)out_size; (void)ws_size;

  const float* x    = (const float*)d_in[0];
  const int*   idx0 = (const int*)d_in[1];
  const int*   idx1 = (const int*)d_in[2];
  const int*   idx2 = (const int*)d_in[3];
  const int*   idx3 = (const int*)d_in[4];
  float* out = (float*)d_out;

  // d_out layout (floats): detail0 | detail1 | detail2 | detail3 | means3
  const size_t len0 = (size_t)B_C * N_C * E_C;            // 67,108,864
  const size_t len1 = len0 / 4, len2 = len0 / 16, len3 = len0 / 64;
  const size_t o1 = len0, o2 = o1 + len1, o3 = o2 + len2, o4 = o3 + len3;

  // ping-pong intermediate means in workspace (~84 MB)
  float* meansA = (float*)d_ws;                           // B * N/4  * E
  float* meansB = meansA + (size_t)B_C * (N_C / 4) * E_C; // B * N/16 * E

  // level 0: x is a 268MB one-shot stream -> NT loads; means0 stays hot in L2
  decomp_level_kernel<true><<<blocks_for((long)B_C * (N_C / 4)),   TPB, 0, stream>>>(
      x,      idx0, out,      meansA,   N_C,       N_C / 4);
  // levels 1-3: sources are the L2-resident means of the previous level
  decomp_level_kernel<false><<<blocks_for((long)B_C * (N_C / 16)), TPB, 0, stream>>>(
      meansA, idx1, out + o1, meansB,   N_C / 4,   N_C / 16);
  decomp_level_kernel<false><<<blocks_for((long)B_C * (N_C / 64)), TPB, 0, stream>>>(
      meansB, idx2, out + o2, meansA,   N_C / 16,  N_C / 64);
  decomp_level_kernel<false><<<blocks_for((long)B_C * (N_C / 256)),TPB, 0, stream>>>(
      meansA, idx3, out + o3, out + o4, N_C / 64,  N_C / 256);
}